// ABC_2D_Large_74182675137249
// MI455X (gfx1250) — compile-verified
//
#include <hip/hip_runtime.h>

typedef __attribute__((ext_vector_type(2))) float v2f;
typedef __attribute__((ext_vector_type(8))) float v8f;

#define B_   16
#define C_   16
#define H_   64
#define W_   64
#define KL_  25
#define O_   32
#define K_   (C_ * KL_)      // 400
#define HW_  (H_ * W_)       // 4096
#define NPIX (B_ * HW_)      // 65536

#define THREADS 256
#define WAVES_PER_BLOCK 8          // wave32: 256 threads = 8 waves
#define PIX_PER_BLOCK (WAVES_PER_BLOCK * 16)   // 128 pixels per block

__global__ __launch_bounds__(THREADS)
void abc2d_large_wmma_f32(const float* __restrict__ x,
                          const int*   __restrict__ conv_hash,
                          const float* __restrict__ zerofy,
                          const float* __restrict__ weights,
                          float*       __restrict__ out)
{
    // Stage full weight matrix (32 x 400 f32 = 50 KB) in LDS; WGP has 320 KB.
    __shared__ __align__(16) float sW[O_ * K_];

    const int tid = threadIdx.x;
    #pragma unroll
    for (int i = tid; i < O_ * K_; i += THREADS) sW[i] = weights[i];
    __syncthreads();

    const int lane = tid & 31;
    const int wave = tid >> 5;
    const int half = lane >> 4;   // 0: lanes 0-15, 1: lanes 16-31
    const int l16  = lane & 15;

    // This lane's pixel (B-matrix column / D-matrix column): n = b*HW + hw
    const int n  = blockIdx.x * PIX_PER_BLOCK + wave * 16 + l16;
    const int b  = n >> 12;            // n / HW_
    const int hw = n & (HW_ - 1);
    // conv_hash/zerofy flat layout: ((b*C + c)*HW + hw)*KL + kk
    const int hashBase = b * (C_ * HW_ * KL_) + hw * KL_;

    v8f acc0 = {};   // output rows o = 0..15
    v8f acc1 = {};   // output rows o = 16..31

    // K loop: V_WMMA_F32_16X16X4_F32, K-step = 4, 100 steps.
    #pragma unroll 2
    for (int k0 = 0; k0 < K_; k0 += 4) {
        const int ka = k0 + half * 2;  // this lane's first k slot (A/B layout: K = 2*half + v)

        // ---- B fragment (4x16 f32): masked gather of 2 values for pixel n ----
        v2f bfrag;
        #pragma unroll
        for (int j = 0; j < 2; ++j) {
            const int k  = ka + j;
            const int c  = k / KL_;            // magic-multiply division by 25
            const int kk = k - c * KL_;
            const int hidx = hashBase + c * (HW_ * KL_) + kk;
            // hash/mask streams are single-use: non-temporal so x stays hot in L2
            const int   gidx = __builtin_nontemporal_load(&conv_hash[hidx]);
            const float z    = __builtin_nontemporal_load(&zerofy[hidx]);
            const float v    = x[gidx];        // L2-resident (16 MB of 192 MB)
            bfrag[j] = (z == 1.0f) ? 0.0f : v;
        }

        // ---- A fragments (16x4 f32) from LDS: rows m=l16 and m=l16+16 ----
        const v2f a0 = *(const v2f*)(&sW[l16 * K_ + ka]);
        const v2f a1 = *(const v2f*)(&sW[(l16 + 16) * K_ + ka]);

        // D = A x B + C ; 8 args: (neg_a, A, neg_b, B, c_mod, C, reuse_a, reuse_b)
        acc0 = __builtin_amdgcn_wmma_f32_16x16x4_f32(false, a0, false, bfrag,
                                                     (short)0, acc0, false, false);
        acc1 = __builtin_amdgcn_wmma_f32_16x16x4_f32(false, a1, false, bfrag,
                                                     (short)0, acc1, false, false);
    }

    // ---- Store D: VGPR r holds row m = r + 8*half, column = l16 (this lane's pixel) ----
    // out layout (B,O,H,W): idx = (b*O + o)*HW + hw
    float* outp = out + b * (O_ * HW_) + hw;
    #pragma unroll
    for (int r = 0; r < 8; ++r) {
        const int m = r + 8 * half;
        outp[m * HW_]        = acc0[r];
        outp[(m + 16) * HW_] = acc1[r];
    }
}

extern "C" void kernel_launch(void* const* d_in, const int* in_sizes, int n_in,
                              void* d_out, int out_size, void* d_ws, size_t ws_size,
                              hipStream_t stream) {
    const float* x         = (const float*)d_in[0];
    const int*   conv_hash = (const int*)  d_in[1];
    const float* zerofy    = (const float*)d_in[2];
    const float* weights   = (const float*)d_in[3];
    float*       out       = (float*)d_out;

    const int blocks = NPIX / PIX_PER_BLOCK;   // 65536 / 128 = 512
    abc2d_large_wmma_f32<<<blocks, THREADS, 0, stream>>>(x, conv_hash, zerofy, weights, out);
}